// Varifold_loss_82403242541370
// MI455X (gfx1250) — compile-verified
//
#include <hip/hip_runtime.h>

typedef __attribute__((ext_vector_type(2))) float v2f;
typedef __attribute__((ext_vector_type(8))) float v8f;

#define NFACES 9976
#define FP     9984            // padded to multiple of 16
#define NT     (FP / 16)       // 624 tiles per side
#define NVERT  5023
#define JPW    13              // j-tiles per wave; 624 = 6 * 8 * 13 exactly

// sigma1 = 0.08 ; sigma2 = 0.02 = sigma1/4
// exp(-D2/s1^2) = exp2(-D2 * log2e/s1^2); exp(-D2/s2^2) = e1^16, weight 1/16
#define A2  225.42109893f      // log2(e) / 0.0064
#define W2  0.0625f

#if __has_builtin(__builtin_amdgcn_exp2f)
#define EXP2F(x) __builtin_amdgcn_exp2f(x)
#else
#define EXP2F(x) exp2f(x)
#endif

// ---------------------------------------------------------------------------
// Phase 1: per-face centers / area-scaled normals for each (frame, mesh).
//   Cp[slot][f] = (Cx, Cy, Cz, |C|^2)
//   Mp[slot][f] = (Mx, My, Mz, 0)  with M = N_raw * rsqrt(|N_raw|)
//   => (Mi.Mj)^2 == (Ni.Nj)^2 * Li * Lj   (areas folded into normals)
// slot = frame*2 + mesh (mesh 0 = predictions, 1 = targets). Padded tail -> 0
// (zero M kills the pair's contribution regardless of the Gaussian).
// ---------------------------------------------------------------------------
__global__ __launch_bounds__(256)
void varifold_setup(const float* __restrict__ pred,
                    const float* __restrict__ targ,
                    const int*   __restrict__ faces,
                    float4* __restrict__ Cp,
                    float4* __restrict__ Mp,
                    float*  __restrict__ out)
{
    int gid = blockIdx.x * blockDim.x + threadIdx.x;
    if (gid == 0) out[0] = 0.0f;           // zero the accumulator each call
    if (gid >= 4 * FP) return;

    int slot  = gid / FP;
    int idx   = gid - slot * FP;
    int frame = slot >> 1;
    int mesh  = slot & 1;

    float4 c = {0.f, 0.f, 0.f, 0.f};
    float4 m = {0.f, 0.f, 0.f, 0.f};

    if (idx < NFACES) {
        const float* V = (mesh == 0 ? pred : targ) + (size_t)frame * NVERT * 3;
        int i0 = faces[idx * 3 + 0];
        int i1 = faces[idx * 3 + 1];
        int i2 = faces[idx * 3 + 2];
        float ax = V[i0*3+0], ay = V[i0*3+1], az = V[i0*3+2];
        float bx = V[i1*3+0], by = V[i1*3+1], bz = V[i1*3+2];
        float dx = V[i2*3+0], dy = V[i2*3+1], dz = V[i2*3+2];

        float cx = (ax + bx + dx) * (1.0f / 3.0f);
        float cy = (ay + by + dy) * (1.0f / 3.0f);
        float cz = (az + bz + dz) * (1.0f / 3.0f);

        float e1x = bx - ax, e1y = by - ay, e1z = bz - az;
        float e2x = dx - ax, e2y = dy - ay, e2z = dz - az;
        float nx = 0.5f * (e1y * e2z - e1z * e2y);
        float ny = 0.5f * (e1z * e2x - e1x * e2z);
        float nz = 0.5f * (e1x * e2y - e1y * e2x);

        float L = sqrtf(nx * nx + ny * ny + nz * nz);     // |N_raw| = area
        float rs = (L > 1e-30f) ? rsqrtf(L) : 0.0f;       // M = N_raw / sqrt(L)

        c = make_float4(cx, cy, cz, cx * cx + cy * cy + cz * cz);
        m = make_float4(nx * rs, ny * rs, nz * rs, 0.0f);
    }
    Cp[gid] = c;
    Mp[gid] = m;
}

// ---------------------------------------------------------------------------
// Phase 2: each wave owns one i-tile and loops over JPW consecutive j-tiles.
// A fragments, the 8 |C_i|^2 broadcasts, the wave reduction and the atomic
// are hoisted out of the j-loop; per tile = 2 B-fragment loads + two
// V_WMMA_F32_16X16X4_F32 (Ci.Cj and Mi.Mj, K=3 padded with a zero lane) +
// 8 exp/lane. grid = (NT/(8*JPW), NT, 6); z = frame*3 + pairset
// {0:slf(+1), 1:cst(+1), 2:crs(-2)}.
// ---------------------------------------------------------------------------
__global__ __launch_bounds__(256)
void varifold_tiles(const float4* __restrict__ Cp,
                    const float4* __restrict__ Mp,
                    float* __restrict__ out)
{
    __shared__ float partial[8];

    const int lane = threadIdx.x & 31;
    const int wv   = threadIdx.x >> 5;
    const int it   = blockIdx.y;
    const int jt0  = (blockIdx.x * 8 + wv) * JPW;
    const int z    = blockIdx.z;
    const int frame = z / 3;
    const int ps    = z - frame * 3;           // 0 slf, 1 cst, 2 crs
    const int meshX = (ps == 1) ? 1 : 0;
    const int meshY = (ps == 0) ? 0 : 1;
    const float coef = (ps == 2) ? -2.0f : 1.0f;

    const float4* Xc = Cp + (size_t)(frame * 2 + meshX) * FP;
    const float4* Xm = Mp + (size_t)(frame * 2 + meshX) * FP;
    const float4* Yc = Cp + (size_t)(frame * 2 + meshY) * FP;
    const float4* Ym = Mp + (size_t)(frame * 2 + meshY) * FP;

    const int  lo = lane & 15;
    const bool hi = lane >= 16;
    const int  i0 = it * 16;

    // ---- hoisted A fragments (row tile) -----------------------------------
    // lanes 0-15: rows i0+lo, K={0,1}=(x,y); lanes 16-31: same rows, K={2,3}=(z,0)
    float4 fa = Xc[i0 + lo];
    float4 ga = Xm[i0 + lo];
    v2f ac = { hi ? fa.z : fa.x, hi ? 0.0f : fa.y };
    v2f am = { hi ? ga.z : ga.x, hi ? 0.0f : ga.y };

    // Pre-scaled row terms: bi[v] = -A2 * |C_row|^2, row = v + 8*(lane>=16).
    const int srcbase = hi ? 8 : 0;
    float bi[8];
#pragma unroll
    for (int v = 0; v < 8; ++v)
        bi[v] = -A2 * __shfl(fa.w, srcbase + v, 32);

    float acc = 0.0f;
    v8f zeroc = {};

    for (int jj = 0; jj < JPW; ++jj) {
        int jt = jt0 + jj;
        if (jt >= NT) break;                   // wave-uniform
        int j0 = jt * 16;

        // ---- B fragments (column tile), same K split across lane halves ----
        float4 fb = Yc[j0 + lo];
        float4 gb = Ym[j0 + lo];
        v2f bc = { hi ? fb.z : fb.x, hi ? 0.0f : fb.y };
        v2f bm = { hi ? gb.z : gb.x, hi ? 0.0f : gb.y };

        // dotC[v] = C_i(row).C_j(col);  dotM[v] = M_i.M_j
        v8f dotC = __builtin_amdgcn_wmma_f32_16x16x4_f32(
            false, ac, false, bc, (short)0, zeroc, false, false);
        v8f dotM = __builtin_amdgcn_wmma_f32_16x16x4_f32(
            false, am, false, bm, (short)0, zeroc, false, false);

        const float bj = -A2 * fb.w;           // -A2 * |C_j|^2 (lane's column)

#pragma unroll
        for (int v = 0; v < 8; ++v) {
            // arg = -A2 * (|Ci|^2 + |Cj|^2 - 2 Ci.Cj)  -> one FMA
            float arg = fmaf(2.0f * A2, dotC[v], bi[v] + bj);
            float uv  = dotM[v];
            float e1  = EXP2F(arg);            // exp(-D2/s1^2)
            float t   = e1 * e1;               // e1^16 = exp(-D2/s2^2)
            t *= t; t *= t; t *= t;
            acc += uv * uv * (e1 + W2 * t);
        }
    }

    // ---- wave32 reduction, then one atomic per block ----------------------
    for (int off = 16; off; off >>= 1)
        acc += __shfl_xor(acc, off, 32);
    if (lane == 0) partial[wv] = acc;
    __syncthreads();

    if (threadIdx.x == 0) {
        float s = 0.0f;
        for (int w = 0; w < 8; ++w) s += partial[w];
        // 0.5 = mean over 2 frames; coef = pair-set sign/weight
        atomicAdd(out, 0.5f * coef * s);
    }
}

// ---------------------------------------------------------------------------
extern "C" void kernel_launch(void* const* d_in, const int* in_sizes, int n_in,
                              void* d_out, int out_size, void* d_ws, size_t ws_size,
                              hipStream_t stream)
{
    (void)in_sizes; (void)n_in; (void)out_size; (void)ws_size;

    const float* pred  = (const float*)d_in[0];   // (2, 5023, 3) f32
    const float* targ  = (const float*)d_in[1];   // (2, 5023, 3) f32
    const int*   faces = (const int*)d_in[2];     // (9976, 3) i32
    float* out = (float*)d_out;                   // scalar f32

    float4* Cp = (float4*)d_ws;                   // 4 * FP float4
    float4* Mp = Cp + 4 * FP;                     // 4 * FP float4 (~1.28 MB total)

    const int total = 4 * FP;
    varifold_setup<<<(total + 255) / 256, 256, 0, stream>>>(pred, targ, faces,
                                                            Cp, Mp, out);

    dim3 grid(NT / (8 * JPW), NT, 6);             // (6, 624, 6)
    varifold_tiles<<<grid, 256, 0, stream>>>(Cp, Mp, out);
}